// LightweightGNN_2491081032400
// MI455X (gfx1250) — compile-verified
//
#include <hip/hip_runtime.h>
#include <stdint.h>

// LightweightGNN: out = LayerNorm(A@x + x), A = sorted-COO sparse matrix.
// Row-per-wave CSR-vector kernel. CDNA5 paths used:
//   * global_load_async_to_lds_b128 (ASYNCcnt) double-buffered gather of x rows
//   * s_wait_asynccnt pipelined waits
//   * global_prefetch_b8 on edge stream
//   * scalarized (SALU/SMEM) binary search + edge metadata via readfirstlane
//   * wave32 __shfl_xor butterfly LayerNorm reduction
//   * non-temporal output stores to keep x resident in the 192MB L2

#ifndef __has_builtin
#define __has_builtin(x) 0
#endif

#define AS1 __attribute__((address_space(1)))
#define AS3 __attribute__((address_space(3)))

typedef int   v4i __attribute__((vector_size(16)));   // 4 x i32 = 128 bits
typedef float v4f __attribute__((vector_size(16)));   // 4 x f32 = 128 bits

#if __has_builtin(__builtin_amdgcn_global_load_async_to_lds_b128)
#define HAVE_ASYNC_B128 1
#else
#define HAVE_ASYNC_B128 0
#endif

#if __has_builtin(__builtin_amdgcn_s_wait_asynccnt)
#define WAIT_ASYNC(n) do { __builtin_amdgcn_s_wait_asynccnt(n); \
                           asm volatile("" ::: "memory"); } while (0)
#else
#define WAIT_ASYNC(n) asm volatile("s_wait_asynccnt %0" :: "i"(n) : "memory")
#endif

namespace {
constexpr int kDim    = 128;  // feature dim; 32 lanes x float4
constexpr int kWaves  = 8;    // waves per block (256 threads, wave32)
constexpr int kBatch  = 4;    // gathered rows per pipeline stage
constexpr float kEps  = 1e-5f;
}

// Scalar (wave-uniform) lower bound: first idx with a[idx] >= key.
__device__ __forceinline__ int lower_bound_u(const int* __restrict__ a,
                                             int lo, int hi, int key) {
  while (lo < hi) {
    int mid = (lo + hi) >> 1;
    if (a[mid] < key) lo = mid + 1; else hi = mid;
  }
  return lo;
}

#if HAVE_ASYNC_B128
__device__ __forceinline__ void issue_row_async(const float* __restrict__ x,
                                                int col, float* dst_row, int d4) {
  const float* src = x + (size_t)col * kDim + d4;
  __builtin_amdgcn_global_load_async_to_lds_b128(
      (AS1 v4i*)(uintptr_t)src,
      (AS3 v4i*)(uint32_t)(uintptr_t)(dst_row + d4),  // generic low 32b == LDS offset
      0, 0);
}
#endif

__global__ __launch_bounds__(kWaves * 32)
void gnn_spmv_ln_kernel(const float* __restrict__ x,
                        const int*   __restrict__ erow,   // sorted, int32
                        const int*   __restrict__ ecol,
                        const float* __restrict__ eval_,
                        const float* __restrict__ gamma,
                        const float* __restrict__ beta,
                        float*       __restrict__ out,
                        int n_rows, int n_edges) {
  // 8 waves x 2 pipeline stages x 4 rows x 512B = 32 KB LDS (of 320 KB/WGP)
  __shared__ float stage_all[kWaves][2][kBatch][kDim];

  const int lane = threadIdx.x & 31;
  const int wave = threadIdx.x >> 5;
  const int row  = blockIdx.x * kWaves + wave;
  if (row >= n_rows) return;

  // Wave-uniform values -> force SGPRs so the search & edge metadata use the
  // scalar (SMEM/SALU) pipe instead of serialized vector loads.
  const int row_u = __builtin_amdgcn_readfirstlane(row);
  const int d4 = lane * 4;                       // this lane owns dims [d4, d4+4)

  const int lo = __builtin_amdgcn_readfirstlane(
      lower_bound_u(erow, 0, n_edges, row_u));
  const int hi = __builtin_amdgcn_readfirstlane(
      lower_bound_u(erow, lo, n_edges, row_u + 1));

  float4 acc = make_float4(0.f, 0.f, 0.f, 0.f);
  int e = lo;

#if HAVE_ASYNC_B128
  float* stage0 = &stage_all[wave][0][0][0];
  float* stage1 = &stage_all[wave][1][0][0];

  // --- remainder first (non-pipelined) so all remaining batches are full ---
  const int rem = (hi - lo) & (kBatch - 1);
  if (rem) {
    for (int j = 0; j < rem; ++j)
      issue_row_async(x, ecol[e + j], stage0 + j * kDim, d4);
    WAIT_ASYNC(0);
    for (int j = 0; j < rem; ++j) {
      const float v = eval_[e + j];
      const float4 xr = *(const float4*)(stage0 + j * kDim + d4);
      acc.x = fmaf(v, xr.x, acc.x);
      acc.y = fmaf(v, xr.y, acc.y);
      acc.z = fmaf(v, xr.z, acc.z);
      acc.w = fmaf(v, xr.w, acc.w);
    }
    e += rem;
  }

  // --- full batches of 4, double-buffered through LDS ---
  if (e < hi) {
    float* bufs[2] = { stage0, stage1 };
    int cur = 0;
#pragma unroll
    for (int j = 0; j < kBatch; ++j)           // prologue: batch 0 in flight
      issue_row_async(x, ecol[e + j], bufs[cur] + j * kDim, d4);

    for (; e + kBatch < hi; e += kBatch) {
      __builtin_prefetch(&ecol[e + 64], 0, 1);  // global_prefetch_b8
      __builtin_prefetch(&eval_[e + 64], 0, 1);
#pragma unroll
      for (int j = 0; j < kBatch; ++j)          // issue next batch
        issue_row_async(x, ecol[e + kBatch + j], bufs[cur ^ 1] + j * kDim, d4);
      WAIT_ASYNC(kBatch);                       // oldest 4 landed (in-order)
#pragma unroll
      for (int j = 0; j < kBatch; ++j) {        // consume current batch
        const float v = eval_[e + j];
        const float4 xr = *(const float4*)(bufs[cur] + j * kDim + d4);
        acc.x = fmaf(v, xr.x, acc.x);
        acc.y = fmaf(v, xr.y, acc.y);
        acc.z = fmaf(v, xr.z, acc.z);
        acc.w = fmaf(v, xr.w, acc.w);
      }
      cur ^= 1;
    }
    WAIT_ASYNC(0);                              // drain epilogue batch
#pragma unroll
    for (int j = 0; j < kBatch; ++j) {
      const float v = eval_[e + j];
      const float4 xr = *(const float4*)(bufs[cur] + j * kDim + d4);
      acc.x = fmaf(v, xr.x, acc.x);
      acc.y = fmaf(v, xr.y, acc.y);
      acc.z = fmaf(v, xr.z, acc.z);
      acc.w = fmaf(v, xr.w, acc.w);
    }
  }
#else
  // Fallback: direct register gather, batch of 4 in flight.
  for (; e < hi; e += kBatch) {
    int nb = hi - e; if (nb > kBatch) nb = kBatch;
    float4 xr[kBatch];
    for (int j = 0; j < nb; ++j)
      xr[j] = *(const float4*)(x + (size_t)ecol[e + j] * kDim + d4);
    for (int j = 0; j < nb; ++j) {
      const float v = eval_[e + j];
      acc.x = fmaf(v, xr[j].x, acc.x);
      acc.y = fmaf(v, xr[j].y, acc.y);
      acc.z = fmaf(v, xr[j].z, acc.z);
      acc.w = fmaf(v, xr[j].w, acc.w);
    }
  }
#endif

  // residual: h = A@x + x
  const float4 xs = *(const float4*)(x + (size_t)row_u * kDim + d4);
  float4 h = make_float4(acc.x + xs.x, acc.y + xs.y, acc.z + xs.z, acc.w + xs.w);

  // LayerNorm over 128 dims: wave32 butterfly reduction of sum / sum-of-squares
  float s  = h.x + h.y + h.z + h.w;
  float sq = h.x * h.x + h.y * h.y + h.z * h.z + h.w * h.w;
#pragma unroll
  for (int off = 16; off > 0; off >>= 1) {
    s  += __shfl_xor(s,  off, 32);
    sq += __shfl_xor(sq, off, 32);
  }
  const float mu   = s * (1.0f / kDim);
  const float var  = sq * (1.0f / kDim) - mu * mu;
  const float rstd = rsqrtf(var + kEps);

  const float4 g = *(const float4*)(gamma + d4);
  const float4 b = *(const float4*)(beta  + d4);
  v4f o;                                    // native vector: NT-store compatible
  o[0] = (h.x - mu) * rstd * g.x + b.x;
  o[1] = (h.y - mu) * rstd * g.y + b.y;
  o[2] = (h.z - mu) * rstd * g.z + b.z;
  o[3] = (h.w - mu) * rstd * g.w + b.w;
  // Write-once output: non-temporal so it doesn't evict L2-resident x.
  __builtin_nontemporal_store(o, (v4f*)(out + (size_t)row_u * kDim + d4));
}

extern "C" void kernel_launch(void* const* d_in, const int* in_sizes, int n_in,
                              void* d_out, int out_size, void* d_ws, size_t ws_size,
                              hipStream_t stream) {
  (void)n_in; (void)out_size; (void)d_ws; (void)ws_size;
  const float* x     = (const float*)d_in[0];
  const int*   erow  = (const int*)  d_in[1];
  const int*   ecol  = (const int*)  d_in[2];
  const float* eval_ = (const float*)d_in[3];
  const float* gamma = (const float*)d_in[4];
  const float* beta  = (const float*)d_in[5];
  float* out = (float*)d_out;

  const int n_rows  = in_sizes[0] / kDim;   // 100000
  const int n_edges = in_sizes[1];          // 3200000

  dim3 block(kWaves * 32);
  dim3 grid((n_rows + kWaves - 1) / kWaves);
  gnn_spmv_ln_kernel<<<grid, block, 0, stream>>>(x, erow, ecol, eval_, gamma,
                                                 beta, out, n_rows, n_edges);
}